// dilate_48799418417409
// MI455X (gfx1250) — compile-verified
//
#include <hip/hip_runtime.h>

#define BATCH 8
#define HGT 128
#define WID 128
#define CHN 64
#define NF 4
#define NPIX (BATCH*HGT*WID)
#define PATCH 576
#define RANGE_C 80.0f
#define LOG2E_F 1.44269504088896340736f
#define LN2_F   0.69314718055994530942f

typedef __attribute__((ext_vector_type(2))) float v2f;
typedef __attribute__((ext_vector_type(8))) float v8f;

__device__ __forceinline__ unsigned fkey(float f) {
  unsigned u = __float_as_uint(f);
  return (u & 0x80000000u) ? ~u : (u | 0x80000000u);
}
__device__ __forceinline__ float funkey(unsigned k) {
  unsigned u = (k & 0x80000000u) ? (k & 0x7FFFFFFFu) : ~k;
  return __uint_as_float(u);
}

__global__ void lse_init(unsigned* __restrict__ wsk) {
  int t = threadIdx.x;
  if (t < CHN) { wsk[t] = 0xFFFFFFFFu; wsk[CHN + t] = 0u; }
}

// Per-channel min/max of x over all B*H*W positions (memory-bound, one 32MB pass).
__global__ __launch_bounds__(256) void lse_chan_minmax(const float* __restrict__ x,
                                                       unsigned* __restrict__ wsk) {
  __shared__ float smn0[256], smx0[256], smn1[256], smx1[256];
  const int t = threadIdx.x;
  const int lane = t & 31;
  const int wv = t >> 5;
  const float INF = __builtin_inff();
  float mn0 = INF, mx0 = -INF, mn1 = INF, mx1 = -INF;
  const float* xp = x + 2 * lane;            // lane owns channels 2*lane, 2*lane+1
  for (long p = (long)blockIdx.x * 8 + wv; p < NPIX; p += (long)gridDim.x * 8) {
    v2f v = *(const v2f*)(xp + p * CHN);
    mn0 = fminf(mn0, v.x); mx0 = fmaxf(mx0, v.x);
    mn1 = fminf(mn1, v.y); mx1 = fmaxf(mx1, v.y);
  }
  smn0[t] = mn0; smx0[t] = mx0; smn1[t] = mn1; smx1[t] = mx1;
  __syncthreads();
  if (t < 32) {
    for (int i = 1; i < 8; ++i) {
      int j = t + 32 * i;
      mn0 = fminf(mn0, smn0[j]); mx0 = fmaxf(mx0, smx0[j]);
      mn1 = fminf(mn1, smn1[j]); mx1 = fmaxf(mx1, smx1[j]);
    }
    atomicMin(&wsk[2*lane],       fkey(mn0));
    atomicMin(&wsk[2*lane+1],     fkey(mn1));
    atomicMax(&wsk[CHN+2*lane],   fkey(mx0));
    atomicMax(&wsk[CHN+2*lane+1], fkey(mx1));
  }
}

// Combine channel extremes with weights + bias extremes -> per-filter LSE params.
__global__ __launch_bounds__(256) void lse_params(const float* __restrict__ kern,
                                                  const float* __restrict__ bias,
                                                  unsigned* __restrict__ wsk) {
  __shared__ float cmn[CHN], cmx[CHN];
  __shared__ float red[256];
  __shared__ float fmn[NF][64], fmx[NF][64];
  const int t = threadIdx.x;
  const float INF = __builtin_inff();
  if (t < CHN) { cmn[t] = funkey(wsk[t]); cmx[t] = funkey(wsk[CHN + t]); }
  float bmn = INF, bmx = -INF;
  for (int i = t; i < HGT * WID; i += 256) {
    float b = bias[i];
    bmn = fminf(bmn, b); bmx = fmaxf(bmx, b);
  }
  red[t] = bmn; __syncthreads();
  for (int s = 128; s > 0; s >>= 1) { if (t < s) red[t] = fminf(red[t], red[t + s]); __syncthreads(); }
  bmn = red[0]; __syncthreads();
  red[t] = bmx; __syncthreads();
  for (int s = 128; s > 0; s >>= 1) { if (t < s) red[t] = fmaxf(red[t], red[t + s]); __syncthreads(); }
  bmx = red[0];
  // 64 threads per filter; element i=slot+64k always has channel==slot
  const int f = t >> 6, slot = t & 63;
  float pmn = INF, pmx = -INF;
  for (int i = slot; i < PATCH; i += 64) {
    float w = kern[f * PATCH + i];
    float c0 = w * cmn[slot], c1 = w * cmx[slot];
    pmn = fminf(pmn, fminf(c0, c1));
    pmx = fmaxf(pmx, fmaxf(c0, c1));
  }
  fmn[f][slot] = pmn; fmx[f][slot] = pmx;
  __syncthreads();
  if (t < NF) {
    float pm = INF, px = -INF;
    for (int i = 0; i < 64; ++i) { pm = fminf(pm, fmn[t][i]); px = fmaxf(px, fmx[t][i]); }
    float wmn = fminf(bmn, pm + bmn);   // zero-padded patch elements contribute bias alone
    float wmx = fmaxf(bmx, px + bmx);
    float range = fmaxf(wmx - wmn, 1e-30f);
    float* wp = (float*)wsk;
    wp[128 + t * 4 + 0] = (RANGE_C / range) * LOG2E_F;  // a: exp2 argument scale
    wp[128 + t * 4 + 1] = wmn;
    wp[128 + t * 4 + 2] = (range / RANGE_C) * LN2_F;    // C1: log2 -> output scale
    wp[128 + t * 4 + 3] = 0.0f;
  }
}

// Main pass: per-pixel logsumexp over 576 patch elems x 4 filters.
// Wave owns 16 pixels as WMMA rows; exp2 on trans pipe, FMA on VALU,
// accumulation on matrix pipe via V_WMMA_F32_16X16X4_F32 with B = ones.
__global__ __launch_bounds__(256) void lse_main(const float* __restrict__ x,
                                                const float* __restrict__ kern,
                                                const float* __restrict__ bias,
                                                const float* __restrict__ wsp,
                                                float* __restrict__ out) {
  __shared__ float sw[NF * PATCH];   // a_f * kernel[f][p]
  __shared__ float prm[16];
  const int t = threadIdx.x;
  if (t < 16) prm[t] = wsp[128 + t];
  for (int i = t; i < NF * PATCH; i += 256) {
    int f = i / PATCH;
    sw[i] = kern[i] * wsp[128 + f * 4];
  }
  __syncthreads();

  const int lane = t & 31;
  const int wv = t >> 5;
  const int grp = blockIdx.x * 8 + wv;  // 16-pixel group id
  const int pix = lane & 15;            // matrix row M (pixel)
  const int half = lane >> 4;           // 0 -> K=0,1 ; 1 -> K=2,3
  const int p = (grp << 4) + pix;
  const int w = p & (WID - 1);
  const int h = (p >> 7) & (HGT - 1);
  const int b = p >> 14;

  const float bi = bias[h * WID + w];
  const float t0 = (bi - prm[1])  * prm[0];
  const float t1 = (bi - prm[5])  * prm[4];
  const float t2 = (bi - prm[9])  * prm[8];
  const float t3 = (bi - prm[13]) * prm[12];

  v8f acc0 = {}, acc1 = {}, acc2 = {}, acc3 = {};
  v2f ones; ones.x = 1.0f; ones.y = 1.0f;

  for (int tap = 0; tap < 9; ++tap) {
    const int dh = tap / 3 - 1, dw = tap % 3 - 1;
    const int hh = h + dh, ww = w + dw;
    const bool valid = (hh >= 0) && (hh < HGT) && (ww >= 0) && (ww < WID);
    const int hc = hh < 0 ? 0 : (hh > HGT - 1 ? HGT - 1 : hh);
    const int wc = ww < 0 ? 0 : (ww > WID - 1 ? WID - 1 : ww);
    const float* xb  = x + ((long)(b * HGT + hc) * WID + wc) * CHN + 2 * half;
    const float* s0b = &sw[0 * PATCH + tap * CHN + 2 * half];
    const float* s1b = &sw[1 * PATCH + tap * CHN + 2 * half];
    const float* s2b = &sw[2 * PATCH + tap * CHN + 2 * half];
    const float* s3b = &sw[3 * PATCH + tap * CHN + 2 * half];
#pragma unroll
    for (int gc = 0; gc < 16; ++gc) {   // 4 channels per group (2 per lane-half)
      v2f xv = *(const v2f*)(xb + gc * 4);
      float x0 = valid ? xv.x : 0.0f;
      float x1 = valid ? xv.y : 0.0f;
      v2f a;
      v2f s0 = *(const v2f*)(s0b + gc * 4);
      a.x = __builtin_amdgcn_exp2f(fmaf(s0.x, x0, t0));
      a.y = __builtin_amdgcn_exp2f(fmaf(s0.y, x1, t0));
      acc0 = __builtin_amdgcn_wmma_f32_16x16x4_f32(false, a, false, ones, (short)0, acc0, false, false);
      v2f s1 = *(const v2f*)(s1b + gc * 4);
      a.x = __builtin_amdgcn_exp2f(fmaf(s1.x, x0, t1));
      a.y = __builtin_amdgcn_exp2f(fmaf(s1.y, x1, t1));
      acc1 = __builtin_amdgcn_wmma_f32_16x16x4_f32(false, a, false, ones, (short)0, acc1, false, false);
      v2f s2 = *(const v2f*)(s2b + gc * 4);
      a.x = __builtin_amdgcn_exp2f(fmaf(s2.x, x0, t2));
      a.y = __builtin_amdgcn_exp2f(fmaf(s2.y, x1, t2));
      acc2 = __builtin_amdgcn_wmma_f32_16x16x4_f32(false, a, false, ones, (short)0, acc2, false, false);
      v2f s3 = *(const v2f*)(s3b + gc * 4);
      a.x = __builtin_amdgcn_exp2f(fmaf(s3.x, x0, t3));
      a.y = __builtin_amdgcn_exp2f(fmaf(s3.y, x1, t3));
      acc3 = __builtin_amdgcn_wmma_f32_16x16x4_f32(false, a, false, ones, (short)0, acc3, false, false);
    }
  }

  // Extract row sums: lane q holds C/D row M=(q&7)+(q>=16?8:0); every column equal.
  const int r = lane & 7;
  float s0v = acc0[0], s1v = acc1[0], s2v = acc2[0], s3v = acc3[0];
#pragma unroll
  for (int i = 1; i < 8; ++i) {
    s0v = (r == i) ? acc0[i] : s0v;
    s1v = (r == i) ? acc1[i] : s1v;
    s2v = (r == i) ? acc2[i] : s2v;
    s3v = (r == i) ? acc3[i] : s3v;
  }
  const int src = (lane & 8) ? (lane + 8) : lane;  // rows 8..15 live in hi half-lanes
  s0v = __shfl(s0v, src, 32);
  s1v = __shfl(s1v, src, 32);
  s2v = __shfl(s2v, src, 32);
  s3v = __shfl(s3v, src, 32);
  if (lane < 16) {
    float4 o;
    o.x = prm[1]  + prm[2]  * __builtin_amdgcn_logf(s0v);
    o.y = prm[5]  + prm[6]  * __builtin_amdgcn_logf(s1v);
    o.z = prm[9]  + prm[10] * __builtin_amdgcn_logf(s2v);
    o.w = prm[13] + prm[14] * __builtin_amdgcn_logf(s3v);
    *(float4*)(out + (long)p * NF) = o;
  }
}

extern "C" void kernel_launch(void* const* d_in, const int* in_sizes, int n_in,
                              void* d_out, int out_size, void* d_ws, size_t ws_size,
                              hipStream_t stream) {
  const float* x    = (const float*)d_in[0];
  const float* kern = (const float*)d_in[1];
  const float* bias = (const float*)d_in[2];
  float* out = (float*)d_out;
  unsigned* wsk = (unsigned*)d_ws;
  lse_init<<<1, 128, 0, stream>>>(wsk);
  lse_chan_minmax<<<256, 256, 0, stream>>>(x, wsk);
  lse_params<<<1, 256, 0, stream>>>(kern, bias, wsk);
  lse_main<<<NPIX / 128, 256, 0, stream>>>(x, kern, bias, (const float*)d_ws, out);
}